// FusedTensorProductOp3_4621384810535
// MI455X (gfx1250) — compile-verified
//
#include <hip/hip_runtime.h>

// FusedTensorProductOp3 for MI455X (gfx1250, wave32).
//
// out[b, s2, u, k] = sum_{p: idx2[p]==s2} sum_{i,j}
//     x0[b, idx0[p], u, i] * x1[b, idx1[p], j] * coeff[p,i,j,k]
//
// Reformulated per batch element b as a dense GEMM:
//     OUT(128 x 96) = X0(128 x 96) * G(96 x 96)
// where column index c = seg*3 + comp, and G aggregates all paths
// (gathers on idx0/idx1, coeff contraction, and the idx2 segment-sum).
// G is built in LDS with ds_add_f32 atomics, then the GEMM runs on
// V_WMMA_F32_16X16X4_F32 (f32 precision == reference precision).

typedef __attribute__((ext_vector_type(2))) float v2f;
typedef __attribute__((ext_vector_type(8))) float v8f;

#define NUM_SEG   32
#define MUL       128
#define D         3
#define NUM_PATHS 64
#define COLS      (NUM_SEG * D)          // 96
#define GPITCH    (COLS + 1)             // 97: pad to avoid LDS bank conflicts
#define X0_PER_B  (NUM_SEG * MUL * D)    // 12288 floats per batch element

__global__ __launch_bounds__(256)
void fused_tp_wmma_kernel(const float* __restrict__ x0,
                          const float* __restrict__ x1,
                          const float* __restrict__ coeff,
                          const int*   __restrict__ idx0,
                          const int*   __restrict__ idx1,
                          const int*   __restrict__ idx2,
                          float*       __restrict__ out)
{
    // 96 x 96 operator matrix, row = (s0*3+i), col = (s2*3+k), pitch-padded.
    __shared__ float G[COLS * GPITCH];   // ~37.3 KB LDS

    const int b   = blockIdx.x;
    const int tid = threadIdx.x;

    // ---------------- Phase 1: zero G ----------------
    for (int t = tid; t < COLS * GPITCH; t += 256) G[t] = 0.0f;
    __syncthreads();

    // ---------------- Phase 2: scatter per-path 3x3 operators into G -------
    if (tid < NUM_PATHS) {
        const int p  = tid;
        const int s0 = idx0[p];
        const int s1 = idx1[p];
        const int s2 = idx2[p];
        const float xv0 = x1[b * COLS + s1 * D + 0];
        const float xv1 = x1[b * COLS + s1 * D + 1];
        const float xv2 = x1[b * COLS + s1 * D + 2];
        const float* cp = coeff + p * (D * D * D);  // coeff[p, i, j, k]
        #pragma unroll
        for (int i = 0; i < D; ++i) {
            #pragma unroll
            for (int k = 0; k < D; ++k) {
                const float m = xv0 * cp[i * 9 + 0 * 3 + k]
                              + xv1 * cp[i * 9 + 1 * 3 + k]
                              + xv2 * cp[i * 9 + 2 * 3 + k];
                // Paths may collide on the same (s0,s2) block -> LDS atomic.
                atomicAdd(&G[(s0 * D + i) * GPITCH + (s2 * D + k)], m);
            }
        }
    }
    __syncthreads();

    // ---------------- Phase 3: GEMM via V_WMMA_F32_16X16X4_F32 -------------
    // 8 waves; wave w owns M-tile rows [16w, 16w+16), all 6 N-tiles.
    const int wave   = tid >> 5;
    const int lane   = tid & 31;
    const int laneLo = lane & 15;
    const int laneHi = lane >> 4;
    const int u      = wave * 16 + laneLo;   // A-matrix row (MUL dimension)

    const float* x0b = x0  + (size_t)b * X0_PER_B;
    float*       ob  = out + (size_t)b * X0_PER_B;

    v8f acc[6] = {};   // 6 N-tiles of 16x16 f32 accumulators (48 VGPRs)

    for (int k0 = 0; k0 < COLS; k0 += 4) {
        // A fragment (16x4 f32): VGPR v, lane-half h -> K = k0 + v + 2h.
        const int c0 = k0 + 2 * laneHi;   // column for A.x
        const int c1 = c0 + 1;            // column for A.y
        // x0 memory layout per b: (s0, u, i) -> s0*384 + u*3 + i, with c = s0*3+i.
        v2f A;
        A.x = x0b[(c0 / D) * (MUL * D) + u * D + (c0 % D)];
        A.y = x0b[(c1 / D) * (MUL * D) + u * D + (c1 % D)];

        #pragma unroll
        for (int n = 0; n < 6; ++n) {
            // B fragment (4x16 f32): VGPR v, lane-half h -> K-row = k0 + v + 2h,
            // column = n*16 + laneLo.  Row reads are conflict-free (pitch 97).
            v2f Bv;
            Bv.x = G[c0 * GPITCH + n * 16 + laneLo];
            Bv.y = G[c1 * GPITCH + n * 16 + laneLo];
            acc[n] = __builtin_amdgcn_wmma_f32_16x16x4_f32(
                /*neg_a=*/false, A, /*neg_b=*/false, Bv,
                /*c_mod=*/(short)0, acc[n],
                /*reuse_a=*/false, /*reuse_b=*/false);
        }
    }

    // ---------------- Phase 4: store C ----------------
    // C 16x16 f32 layout: VGPR r holds M = r + 8*laneHi, N = laneLo.
    // Output layout per b: (s2, m, k) -> (c2/3)*384 + m*3 + (c2%3), c2 = s2*3+k.
    #pragma unroll
    for (int n = 0; n < 6; ++n) {
        #pragma unroll
        for (int r = 0; r < 8; ++r) {
            const int m  = wave * 16 + r + 8 * laneHi;
            const int c2 = n * 16 + laneLo;
            ob[(c2 / D) * (MUL * D) + m * D + (c2 % D)] = acc[n][r];
        }
    }
}

extern "C" void kernel_launch(void* const* d_in, const int* in_sizes, int n_in,
                              void* d_out, int out_size, void* d_ws, size_t ws_size,
                              hipStream_t stream)
{
    const float* x0    = (const float*)d_in[0];
    const float* x1    = (const float*)d_in[1];
    const float* coeff = (const float*)d_in[2];
    const int*   idx0  = (const int*)d_in[3];
    const int*   idx1  = (const int*)d_in[4];
    const int*   idx2  = (const int*)d_in[5];
    float*       out   = (float*)d_out;

    const int batch = in_sizes[0] / X0_PER_B;   // 2048
    fused_tp_wmma_kernel<<<batch, 256, 0, stream>>>(x0, x1, coeff,
                                                    idx0, idx1, idx2, out);
}